// RetinaFace_21474836480602
// MI455X (gfx1250) — compile-verified
//
#include <hip/hip_runtime.h>
#include <math.h>

#define N_IMG   32
#define A_NUM   67200
#define TOPK    1024
#define CONF_T  0.5f
#define NMS_T   0.4f
#define IMF     1280.0f

typedef __attribute__((ext_vector_type(2))) float v2f;
typedef __attribute__((ext_vector_type(8))) float v8f;

__device__ __forceinline__ unsigned sortable_key(float f) {
    unsigned u = __float_as_uint(f);
    return (u & 0x80000000u) ? ~u : (u | 0x80000000u);
}

// ---------------------------------------------------------------------------
// 1) softmax face score:  s = exp(c1)/(exp(c0)+exp(c1)) = 1/(1+exp(c0-c1))
// ---------------------------------------------------------------------------
__global__ void score_kernel(const float* __restrict__ conf,
                             float* __restrict__ scores) {
    int i = blockIdx.x * blockDim.x + threadIdx.x;
    if (i < N_IMG * A_NUM) {
        float c0 = conf[2 * i], c1 = conf[2 * i + 1];
        scores[i] = 1.0f / (1.0f + __expf(c0 - c1));
    }
}

// ---------------------------------------------------------------------------
// 2) per-image exact top-K: 4-pass radix select on sortable keys, compaction,
//    then in-LDS bitonic sort (key desc, index asc). One 1024-thread block
//    per image.
// ---------------------------------------------------------------------------
__global__ void __launch_bounds__(1024)
topk_kernel(const float* __restrict__ scores,
            float* __restrict__ topv, int* __restrict__ topi) {
    __shared__ unsigned hist[256];
    __shared__ unsigned sPrefix, sRemain, sCntHi, sCntEq;
    __shared__ unsigned skey[TOPK];
    __shared__ int      sidx[TOPK];

    const int n = blockIdx.x, t = threadIdx.x;
    const float* s = scores + (size_t)n * A_NUM;

    if (t == 0) { sPrefix = 0u; sRemain = TOPK; }
    __syncthreads();

    // ---- radix select (MSB -> LSB byte) for exact K-th largest key ----
    for (int pass = 0; pass < 4; ++pass) {
        const int shift = 24 - 8 * pass;
        if (t < 256) hist[t] = 0u;
        __syncthreads();
        const unsigned pfx = sPrefix;
        for (int a = t; a < A_NUM; a += 1024) {
            unsigned key = sortable_key(s[a]);
            bool qual = (pass == 0) ||
                (((unsigned long long)(key ^ pfx) >> (unsigned)(shift + 8)) == 0ull);
            if (qual) atomicAdd(&hist[(key >> shift) & 255u], 1u);
        }
        __syncthreads();
        if (t == 0) {
            unsigned R = sRemain, cum = 0u;
            for (int b = 255; b >= 0; --b) {
                unsigned c = hist[b];
                if (cum + c >= R) { sPrefix = pfx | ((unsigned)b << shift);
                                    sRemain = R - cum; break; }
                cum += c;
            }
        }
        __syncthreads();
    }

    const unsigned T = sPrefix;        // key of the K-th largest element
    const unsigned needEq = sRemain;   // how many ties at T we take
    if (t == 0) { sCntHi = 0u; sCntEq = 0u; }
    __syncthreads();

    // strictly-greater candidates (exactly TOPK - needEq of them)
    for (int a = t; a < A_NUM; a += 1024) {
        unsigned key = sortable_key(s[a]);
        if (key > T) {
            unsigned p = atomicAdd(&sCntHi, 1u);
            skey[p] = key; sidx[p] = a;
        }
    }
    __syncthreads();
    const unsigned base = sCntHi;
    for (int a = t; a < A_NUM; a += 1024) {
        unsigned key = sortable_key(s[a]);
        if (key == T) {
            unsigned q = atomicAdd(&sCntEq, 1u);
            if (q < needEq) { skey[base + q] = key; sidx[base + q] = a; }
        }
    }
    __syncthreads();

    // ---- bitonic sort, rank order = (key desc, idx asc) ----
    for (int k = 2; k <= TOPK; k <<= 1) {
        for (int j = k >> 1; j > 0; j >>= 1) {
            int ixj = t ^ j;
            if (ixj > t) {
                unsigned ka = skey[t], kb = skey[ixj];
                int ia = sidx[t], ib = sidx[ixj];
                bool aAfterB = (ka < kb) || (ka == kb && ia > ib);
                bool doSwap = ((t & k) == 0) ? aAfterB : !aAfterB;
                if (doSwap) { skey[t] = kb; skey[ixj] = ka;
                              sidx[t] = ib; sidx[ixj] = ia; }
            }
            __syncthreads();
        }
    }

    unsigned key = skey[t];
    unsigned u = (key & 0x80000000u) ? (key & 0x7FFFFFFFu) : ~key;
    topv[n * TOPK + t] = __uint_as_float(u);
    topi[n * TOPK + t] = sidx[t];
}

// ---------------------------------------------------------------------------
// 3) gather + decode boxes / landmarks for the K candidates only
// ---------------------------------------------------------------------------
__global__ void decode_kernel(const float* __restrict__ loc,
                              const float* __restrict__ landms,
                              const float* __restrict__ priors,
                              const int* __restrict__ topi,
                              float* __restrict__ boxes,
                              float* __restrict__ pts) {
    int g = blockIdx.x * blockDim.x + threadIdx.x;
    if (g >= N_IMG * TOPK) return;
    int n = g / TOPK;
    int i = topi[g];

    float4 pr = ((const float4*)priors)[i];                  // cx,cy,w,h
    float4 lc = ((const float4*)loc)[(size_t)n * A_NUM + i];

    float cx = pr.x + lc.x * 0.1f * pr.z;
    float cy = pr.y + lc.y * 0.1f * pr.w;
    float w  = pr.z * __expf(lc.z * 0.2f);
    float h  = pr.w * __expf(lc.w * 0.2f);
    float4 b;
    b.x = (cx - 0.5f * w) * IMF;  b.y = (cy - 0.5f * h) * IMF;
    b.z = (cx + 0.5f * w) * IMF;  b.w = (cy + 0.5f * h) * IMF;
    ((float4*)boxes)[g] = b;

    const float* lm = landms + ((size_t)n * A_NUM + i) * 10;
    float* po = pts + (size_t)g * 10;
#pragma unroll
    for (int p = 0; p < 5; ++p) {
        po[2 * p]     = (pr.x + lm[2 * p]     * 0.1f * pr.z) * IMF;
        po[2 * p + 1] = (pr.y + lm[2 * p + 1] * 0.1f * pr.w) * IMF;
    }
}

// ---------------------------------------------------------------------------
// 4) NMS per image (one 1024-thread block):
//    - async-stage 1024 boxes (16KB) global -> LDS (gfx1250 async path)
//    - suppression bitmatrix in LDS (1024x1024 bits = 128KB), built in 16x16
//      tiles: area-sum via V_WMMA_F32_16X16X4_F32 rank-2 outer product,
//      intersections on VALU, bits packed with wave32 ballots
//    - greedy scan by wave 0 using shuffles (no barriers in the scan loop)
//    - fused masked det[N,K,15] output
// ---------------------------------------------------------------------------
__global__ void __launch_bounds__(1024)
nms_kernel(const float* __restrict__ boxes,
           const float* __restrict__ pts,
           const float* __restrict__ topv,
           float* __restrict__ det) {
    extern __shared__ unsigned smem[];
    unsigned* sSup  = smem;                                  // 1024*32 u32 (128KB)
    float*    sBoxF = (float*)(smem + TOPK * 32);            // 1024*4  f32 (16KB)
    float*    sArea = sBoxF + TOPK * 4;                      // 1024    f32 (4KB)
    unsigned* sKeep = (unsigned*)(sArea + TOPK);             // 32      u32

    const int n = blockIdx.x, t = threadIdx.x;
    const int wave = t >> 5, lane = t & 31;

    // --- issue async global->LDS copy of this image's candidate boxes ---
    {
        const float* gsrc = boxes + (size_t)n * TOPK * 4;
        unsigned ldsDst = (unsigned)(size_t)&sBoxF[t * 4];
        unsigned voff   = (unsigned)(t * 16);
        asm volatile("global_load_async_to_lds_b128 %0, %1, %2"
                     :: "v"(ldsDst), "v"(voff), "s"(gsrc) : "memory");
    }
    // overlap: zero the suppression bitmatrix while the DMA is in flight
    for (int k = t; k < TOPK * 32; k += 1024) sSup[k] = 0u;
    asm volatile("s_wait_asynccnt 0" ::: "memory");
    __syncthreads();

    const float4* sBox = (const float4*)sBoxF;
    {
        float4 b = sBox[t];
        sArea[t] = fmaxf(b.z - b.x, 0.0f) * fmaxf(b.w - b.y, 0.0f);
    }
    __syncthreads();

    // --- suppression matrix, upper triangle in 16x16 tiles, load-balanced:
    //     wave w owns row-tiles {w, 63-w} (65 tiles each) ---
    const int rts[2] = { wave, 63 - wave };
    for (int r = 0; r < 2; ++r) {
        const int rt = rts[r];
        v2f afrag;                      // A: 16x4 = [area_i, 1, 0, 0]
        afrag.x = (lane < 16) ? sArea[rt * 16 + lane] : 0.0f;
        afrag.y = (lane < 16) ? 1.0f : 0.0f;
        for (int ct = rt; ct < 64; ++ct) {
            v2f bfrag;                  // B: 4x16 = [[1..1],[area_j..],0,0]
            bfrag.x = (lane < 16) ? 1.0f : sArea[ct * 16 + (lane - 16)];
            bfrag.y = 0.0f;
            v8f c = {0.f, 0.f, 0.f, 0.f, 0.f, 0.f, 0.f, 0.f};
            // D[m][n] = area_i[m] + area_j[n]  (union base term)
            c = __builtin_amdgcn_wmma_f32_16x16x4_f32(
                    false, afrag, false, bfrag, (short)0, c, false, false);

            const int nn = lane & 15;
            const int mBase = (lane < 16) ? 0 : 8;
            const unsigned sh = (unsigned)(ct & 1) * 16u;
#pragma unroll
            for (int v = 0; v < 8; ++v) {
                int gi = rt * 16 + mBase + v;
                int gj = ct * 16 + nn;
                float4 bi = sBox[gi], bj = sBox[gj];
                float ix = fminf(bi.z, bj.z) - fmaxf(bi.x, bj.x);
                float iy = fminf(bi.w, bj.w) - fmaxf(bi.y, bj.y);
                float inter = fmaxf(ix, 0.0f) * fmaxf(iy, 0.0f);
                float uni = c[v] - inter + 1e-9f;
                bool sup = (inter > NMS_T * uni) && (gj > gi);
                unsigned mask = (unsigned)__ballot(sup);
                if (lane == v)
                    atomicOr(&sSup[(rt * 16 + v) * 32 + (ct >> 1)],
                             (mask & 0xFFFFu) << sh);
                if (lane == v + 8)
                    atomicOr(&sSup[(rt * 16 + 8 + v) * 32 + (ct >> 1)],
                             (mask >> 16) << sh);
            }
        }
    }

    // --- keep-mask init: score > CONF_T ---
    float sc = topv[n * TOPK + t];
    unsigned vm = (unsigned)__ballot(sc > CONF_T);
    if (lane == 0) sKeep[wave] = vm;
    __syncthreads();

    // --- greedy scan: wave 0, lane l owns keep-word l; shuffle broadcasts ---
    if (wave == 0) {
        unsigned keepW = sKeep[lane];
        for (int i = 0; i < TOPK; ++i) {
            unsigned ow = __shfl(keepW, i >> 5, 32);
            if ((ow >> (i & 31)) & 1u)
                keepW &= ~sSup[i * 32 + lane];
        }
        sKeep[lane] = keepW;
    }
    __syncthreads();

    // --- fused masked output: det[n,t,0:15] = [box4, score, lm10] * keep ---
    float kf = ((sKeep[wave] >> lane) & 1u) ? 1.0f : 0.0f;
    float4 b = sBox[t];
    float* o = det + (size_t)(n * TOPK + t) * 15;
    const float* pp = pts + (size_t)(n * TOPK + t) * 10;
    o[0] = b.x * kf; o[1] = b.y * kf; o[2] = b.z * kf; o[3] = b.w * kf;
    o[4] = sc * kf;
#pragma unroll
    for (int p = 0; p < 10; ++p) o[5 + p] = pp[p] * kf;
}

// ---------------------------------------------------------------------------
extern "C" void kernel_launch(void* const* d_in, const int* in_sizes, int n_in,
                              void* d_out, int out_size, void* d_ws, size_t ws_size,
                              hipStream_t stream) {
    const float* loc    = (const float*)d_in[0];
    const float* conf   = (const float*)d_in[1];
    const float* landms = (const float*)d_in[2];
    const float* priors = (const float*)d_in[3];
    float* det = (float*)d_out;

    // workspace layout (all 16B aligned)
    char* ws = (char*)d_ws;
    constexpr size_t SCORES_B = (size_t)N_IMG * A_NUM * 4;          // 8.6 MB
    constexpr size_t TOPV_B   = (size_t)N_IMG * TOPK * 4;
    constexpr size_t TOPI_B   = (size_t)N_IMG * TOPK * 4;
    constexpr size_t BOXES_B  = (size_t)N_IMG * TOPK * 16;
    float* scores = (float*)(ws);
    float* topv   = (float*)(ws + SCORES_B);
    int*   topi   = (int*)  (ws + SCORES_B + TOPV_B);
    float* boxes  = (float*)(ws + SCORES_B + TOPV_B + TOPI_B);
    float* pts    = (float*)(ws + SCORES_B + TOPV_B + TOPI_B + BOXES_B);

    int total = N_IMG * A_NUM;
    score_kernel<<<(total + 255) / 256, 256, 0, stream>>>(conf, scores);
    topk_kernel<<<N_IMG, 1024, 0, stream>>>(scores, topv, topi);
    decode_kernel<<<(N_IMG * TOPK + 255) / 256, 256, 0, stream>>>(
        loc, landms, priors, topi, boxes, pts);

    // dynamic LDS: sup bitmatrix 128KB + boxes 16KB + areas 4KB + keep 128B
    size_t shmem = (size_t)TOPK * 32 * 4 + (size_t)TOPK * 4 * 4
                 + (size_t)TOPK * 4 + 32 * 4;
    nms_kernel<<<N_IMG, 1024, shmem, stream>>>(boxes, pts, topv, det);
}